// YOLOLayer1_20641612824960
// MI455X (gfx1250) — compile-verified
//
#include <hip/hip_runtime.h>
#include <stdint.h>

// ---------------------------------------------------------------------------
// YOLO layer post-process for MI455X (gfx1250).
// TDM stages the channel-major input tile into LDS (hardware pad => 257-float
// pitch, bank-conflict-free transposed reads). Phase 2 splits work:
//   - 5 box/conf channels: specialized sigmoid math (compile-time channel)
//   - 60 class channels:   pure LDS->global copy, coalesced NT stores
// keeping the kernel under the HBM roofline instead of issue-bound.
// ---------------------------------------------------------------------------

typedef uint32_t v4u32 __attribute__((ext_vector_type(4)));
typedef int      v8i32 __attribute__((ext_vector_type(8)));
typedef int      v4i32 __attribute__((ext_vector_type(4)));

#define NG     128
#define SPAT   (NG * NG)      // 16384 spatial positions per (b, a)
#define NA     4
#define CCH    65             // 5 + 60 channels
#define NCLS   60
#define NCHAN  (NA * CCH)     // 260
#define TILE   256            // spatial positions per workgroup
#define TILEP  257            // LDS pitch (pad 1 dword per 256 via TDM padding)

__device__ __forceinline__ float sigmoidf_fast(float x) {
    return __builtin_amdgcn_rcpf(1.0f + __expf(-x));   // v_rcp_f32 / v_exp_f32
}

__global__ __launch_bounds__(256)
void yolo_tdm_kernel(const float* __restrict__ p,
                     const float* __restrict__ anchors,
                     const int*   __restrict__ img_dim,
                     float*       __restrict__ out)
{
    __shared__ float tile[CCH * TILEP];   // 65 * 257 * 4 = 66,820 bytes

    const int t  = threadIdx.x;
    const int ti = blockIdx.x;            // spatial tile index 0..63
    const int a  = blockIdx.y;            // anchor 0..3
    const int b  = blockIdx.z;            // batch
    const int s0 = ti * TILE;

    const float* src = p + ((size_t)(b * NCHAN + a * CCH) * SPAT + s0);

    // ---- Phase 1: TDM DMA of a 65 x 256 f32 tile (row stride 16384 elems)
    // into LDS with 1-dword padding after every 256 dwords (pitch 257).
    // TDM ignores EXEC; issue from wave 0 only, then wait + barrier.
    if (t < 32) {
        const uint64_t gaddr    = (uint64_t)(uintptr_t)src;
        const uint32_t lds_base =
            (uint32_t)(uintptr_t)(__attribute__((address_space(3))) float*)tile;

        v4u32 g0;
        g0.x = 1u;                                    // count=1, is_restore=0
        g0.y = lds_base;                              // lds_addr [63:32]
        g0.z = (uint32_t)gaddr;                       // global_addr lo
        g0.w = (uint32_t)(gaddr >> 32) | (2u << 30);  // global_addr hi | type=2

        v8i32 g1;
        // data_size=2 (4B) | pad_enable | pad_interval=7 (256 dw) | pad_amount=0 (1 dw)
        g1[0] = (int)(0x00020000u | 0x00100000u | (7u << 22));
        g1[1] = (int)((SPAT & 0xFFFF) << 16);         // tensor_dim0 lo16 (16384)
        g1[2] = (int)(((SPAT >> 16) & 0xFFFF) | (CCH << 16)); // dim0 hi16 | dim1 lo16
        g1[3] = (int)(TILE << 16);                    // dim1 hi16=0 | tile_dim0=256
        g1[4] = (int)CCH;                             // tile_dim1=65 | tile_dim2=0
        g1[5] = (int)SPAT;                            // tensor_dim0_stride lo32
        g1[6] = 0;                                    // stride hi | dim1_stride lo
        g1[7] = 0;                                    // dim1_stride hi (unused, 2D)

        v4i32 gz4 = {0, 0, 0, 0};                     // groups 2/3 unused (2D tile)
        v8i32 gz8 = {0, 0, 0, 0, 0, 0, 0, 0};        // extra group (clang-23 form)

        __builtin_amdgcn_tensor_load_to_lds(g0, g1, gz4, gz4, gz8, 0);
        __builtin_amdgcn_s_wait_tensorcnt(0);
    }
    __syncthreads();
    // TDM wrote LDS behind the compiler's back (descriptor is by-value):
    // a memory-clobber barrier keeps the tile[] loads live and ordered.
    asm volatile("" ::: "memory");

    float* outp = out + ((size_t)(b * NA + a) * SPAT + s0) * (size_t)CCH;

    // ---- Phase 2a: box/conf channels (c = 0..4), spatial position sp = t.
    // Channel is a compile-time constant per statement: no selects, no masks.
    {
        const float aw   = anchors[a * 2 + 0];
        const float ah   = anchors[a * 2 + 1];
        const float strd = (float)img_dim[0] / (float)NG;   // 1024/128 = 8.0
        const int   g    = s0 + t;
        const float gx   = (float)(g & (NG - 1));
        const float gy   = (float)(g >> 7);

        const float sx = sigmoidf_fast(tile[0 * TILEP + t]);
        const float sy = sigmoidf_fast(tile[1 * TILEP + t]);
        const float sw = sigmoidf_fast(tile[2 * TILEP + t]);
        const float sh = sigmoidf_fast(tile[3 * TILEP + t]);
        const float sc = sigmoidf_fast(tile[4 * TILEP + t]);

        float* row = outp + t * CCH;
        __builtin_nontemporal_store((sx + gx) * strd,      row + 0);  // x
        __builtin_nontemporal_store((sy + gy) * strd,      row + 1);  // y
        __builtin_nontemporal_store(4.0f * sw * sw * aw,   row + 2);  // width
        __builtin_nontemporal_store(4.0f * sh * sh * ah,   row + 3);  // height
        __builtin_nontemporal_store(sc,                    row + 4);  // conf
    }

    // ---- Phase 2b: class channels (c = 5..64): pure transposed copy.
    // Flat job j = t + 256*i over 15360 class elements: sp = j/60, cc = j%60.
    // ds_load lane-stride 257 (conflict-free), stores coalesced (+6 hop/row).
    int sp = t / NCLS;
    int cc = t - sp * NCLS;
#pragma unroll
    for (int i = 0; i < NCLS; ++i) {
        const float v = tile[(cc + 5) * TILEP + sp];
        __builtin_nontemporal_store(v, outp + sp * CCH + cc + 5);

        cc += 256 - 4 * NCLS;             // +16   (256 = 4*60 + 16)
        sp += 4;
        if (cc >= NCLS) { cc -= NCLS; ++sp; }
    }
}

extern "C" void kernel_launch(void* const* d_in, const int* in_sizes, int n_in,
                              void* d_out, int out_size, void* d_ws, size_t ws_size,
                              hipStream_t stream)
{
    const float* p       = (const float*)d_in[0];
    const float* anchors = (const float*)d_in[1];
    const int*   img_dim = (const int*)d_in[2];
    float*       out     = (float*)d_out;

    const int bs = in_sizes[0] / (NCHAN * SPAT);   // 8 for the reference shapes

    dim3 grid(SPAT / TILE, NA, bs);                // 64 x 4 x 8 = 2048 WGs
    dim3 block(256);
    yolo_tdm_kernel<<<grid, block, 0, stream>>>(p, anchors, img_dim, out);
}